// PatchDistillHead_79886391705739
// MI455X (gfx1250) — compile-verified
//
#include <hip/hip_runtime.h>
#include <hip/hip_bf16.h>

// ---------------------------------------------------------------------------
// PatchDistillHead for MI455X (gfx1250):
//   - all heavy math on V_WMMA_F32_16X16X4_F32 (fp32, matches reference dtype)
//   - B tiles staged to LDS via GLOBAL_LOAD_ASYNC_TO_LDS_B128, double-buffered,
//     synchronized with s_wait_asynccnt (CDNA5 async path)
// ---------------------------------------------------------------------------

typedef float v2f __attribute__((ext_vector_type(2)));
typedef float v8f __attribute__((ext_vector_type(8)));

#define NTOK   32768   // 8 images * 4096 tokens
#define TOKPI  4096
#define DIN    1176
#define DMODEL 1024
#define NEXP   8
#define BN     64
#define NLAY   4
#define OUTDIM 4096

#define KS      8            // K-rows staged per async stage
#define BSTRIDE 72           // LDS row stride in floats (16B aligned, bank-skewed)
#define BBUF    (KS * BSTRIDE)

// issue one b128 async global->LDS transfer per thread (128 threads = 2KB stage)
__device__ __forceinline__ void async_stage_B(const float* __restrict__ B, int ldb,
                                              int col0, int k0, float* bs)
{
  const int brow   = threadIdx.x >> 4;   // 0..7
  const int bchunk = threadIdx.x & 15;   // 0..15 (x4 floats = 64 cols)
  const float* src = B + (size_t)(k0 + brow) * ldb + col0 + bchunk * 4;
  const unsigned dst = (unsigned)(size_t)(bs + brow * BSTRIDE + bchunk * 4);
  asm volatile("global_load_async_to_lds_b128 %0, %1, off"
               :: "v"(dst), "v"((unsigned long long)(size_t)src)
               : "memory");
}

// ---------------------------------------------------------------------------
// Wave-level 16x64 fp32 WMMA tile with LDS-staged B:
//   C[row0:+16, col0:+64] += A[row0:+16, :K] @ B[:K, col0:+64]
// A layout (ISA 32-bit A 16x4): lanes 0-15 hold K={k,k+1}, lanes 16-31 K={k+2,k+3}
// bs: shared buffer of 2*KS*BSTRIDE floats (double buffered)
// ---------------------------------------------------------------------------
__device__ __forceinline__ void wmma_tile_16x64(
    const float* __restrict__ A, int lda,
    const float* __restrict__ B, int ldb,
    int K, int row0, int col0, float* bs, v8f c[4])
{
  const int lane = threadIdx.x & 31;
  const int kh   = lane >> 4;   // lane half selects K pair
  const int mn   = lane & 15;   // M for A-frag, N for B-frag
  const float* Arow = A + (size_t)(row0 + mn) * lda;

  async_stage_B(B, ldb, col0, 0, bs);          // prologue prefetch -> buf0

  for (int k0 = 0; k0 < K; k0 += KS) {
    float* cur = bs + ((k0 >> 3) & 1) * BBUF;
    float* nxt = bs + ((((k0 >> 3) & 1) ^ 1)) * BBUF;
    if (k0 + KS < K) {
      async_stage_B(B, ldb, col0, k0 + KS, nxt);
      asm volatile("s_wait_asynccnt 0x1" ::: "memory");  // drain current stage
    } else {
      asm volatile("s_wait_asynccnt 0x0" ::: "memory");
    }
    __syncthreads();
#pragma unroll
    for (int kk = 0; kk < KS; kk += 4) {
      const int ka = k0 + kk + 2 * kh;
      v2f a;
      a.x = Arow[ka];
      a.y = Arow[ka + 1];
      const float* bb = cur + (kk + 2 * kh) * BSTRIDE + mn;
#pragma unroll
      for (int j = 0; j < 4; ++j) {
        v2f b;
        b.x = bb[j * 16];
        b.y = bb[j * 16 + BSTRIDE];
        c[j] = __builtin_amdgcn_wmma_f32_16x16x4_f32(
            false, a, false, b, (short)0, c[j], false, false);
      }
    }
    __syncthreads();  // all waves done reading before buffer is re-filled
  }
}

// ---------------------------------------------------------------------------
// Stage 1: h = X @ w_in   ([32768 x 1176] @ [1176 x 1024])
// ---------------------------------------------------------------------------
__global__ void __launch_bounds__(128)
gemm_in_kernel(const float* __restrict__ X, const float* __restrict__ Wi,
               float* __restrict__ h)
{
  __shared__ __align__(16) float Bs[2 * BBUF];
  const int row0 = blockIdx.y * 64 + (threadIdx.x >> 5) * 16;
  const int col0 = blockIdx.x * 64;
  v8f c[4] = {};
  wmma_tile_16x64(X, DIN, Wi, DMODEL, DIN, row0, col0, Bs, c);
  const int lane = threadIdx.x & 31;
  const int kh = lane >> 4, mn = lane & 15;
#pragma unroll
  for (int j = 0; j < 4; ++j)
#pragma unroll
    for (int r = 0; r < 8; ++r)
      h[(size_t)(row0 + r + 8 * kh) * DMODEL + col0 + j * 16 + mn] = c[j][r];
}

// ---------------------------------------------------------------------------
// MoE GEMM 1: h1w[t, e*64+b] = gate[t,e] * relu( (h @ W1[l,e])[t,b] + b1[l,e,b] )
// Each block covers one expert's full 64 columns (blockIdx.x == e).
// ---------------------------------------------------------------------------
__global__ void __launch_bounds__(128)
gemm1_kernel(const float* __restrict__ h, const float* __restrict__ ew1,
             const float* __restrict__ eb1, const float* __restrict__ wfull,
             float* __restrict__ h1w, int l)
{
  __shared__ __align__(16) float Bs[2 * BBUF];
  const int row0 = blockIdx.y * 64 + (threadIdx.x >> 5) * 16;
  const int e    = blockIdx.x;                                  // expert
  const float* B = ew1 + (size_t)(l * NEXP + e) * DMODEL * BN;  // [1024 x 64]
  v8f c[4] = {};
  wmma_tile_16x64(h, DMODEL, B, BN, DMODEL, row0, 0, Bs, c);
  const int lane = threadIdx.x & 31;
  const int kh = lane >> 4, mn = lane & 15;
#pragma unroll
  for (int j = 0; j < 4; ++j) {
    const int bcol = j * 16 + mn;
    const float bias = eb1[(l * NEXP + e) * BN + bcol];
#pragma unroll
    for (int r = 0; r < 8; ++r) {
      const int t = row0 + r + 8 * kh;
      float v = c[j][r] + bias;
      v = v > 0.f ? v : 0.f;
      v *= wfull[t * NEXP + e];
      h1w[(size_t)t * (NEXP * BN) + e * BN + bcol] = v;
    }
  }
}

// ---------------------------------------------------------------------------
// MoE GEMM 2 + residual: h[t,d] += (h1w @ W2flat)[t,d] + sum_e gate[t,e]*b2[l,e,d]
// W2flat = ew2[l] viewed as contiguous [512 x 1024].
// ---------------------------------------------------------------------------
__global__ void __launch_bounds__(128)
gemm2_kernel(const float* __restrict__ h1w, const float* __restrict__ ew2,
             const float* __restrict__ eb2, const float* __restrict__ wfull,
             float* __restrict__ h, int l)
{
  __shared__ __align__(16) float Bs[2 * BBUF];
  const int row0 = blockIdx.y * 64 + (threadIdx.x >> 5) * 16;
  const int col0 = blockIdx.x * 64;
  const float* B = ew2 + (size_t)l * (NEXP * BN) * DMODEL;
  v8f c[4] = {};
  wmma_tile_16x64(h1w, NEXP * BN, B, DMODEL, NEXP * BN, row0, col0, Bs, c);
  const int lane = threadIdx.x & 31;
  const int kh = lane >> 4, mn = lane & 15;
#pragma unroll
  for (int j = 0; j < 4; ++j) {
    const int d = col0 + j * 16 + mn;
#pragma unroll
    for (int r = 0; r < 8; ++r) {
      const int t = row0 + r + 8 * kh;
      float bias = 0.f;
#pragma unroll
      for (int e = 0; e < NEXP; ++e)
        bias += wfull[t * NEXP + e] * eb2[(size_t)(l * NEXP + e) * DMODEL + d];
      const size_t idx = (size_t)t * DMODEL + d;
      h[idx] = h[idx] + c[j][r] + bias;
    }
  }
}

// ---------------------------------------------------------------------------
// Router: one wave per token. logits = h@w_rimg[l] + skin_probs@w_rskin[l],
// softmax, top-2, renormalize, dense gates; aux sums via atomics.
// ---------------------------------------------------------------------------
__global__ void __launch_bounds__(256)
router_kernel(const float* __restrict__ h, const float* __restrict__ w_rimg,
              const float* __restrict__ w_rskin, const float* __restrict__ sprob,
              float* __restrict__ wfull, float* __restrict__ psum,
              float* __restrict__ msum, int l)
{
  const int t    = (blockIdx.x * blockDim.x + threadIdx.x) >> 5;
  const int lane = threadIdx.x & 31;
  if (t >= NTOK) return;
  const int img = t >> 12;
  const float* hrow = h + (size_t)t * DMODEL;
  const float* wr = w_rimg + (size_t)l * DMODEL * NEXP;
  float acc[NEXP];
#pragma unroll
  for (int e = 0; e < NEXP; ++e) acc[e] = 0.f;
  for (int d = lane; d < DMODEL; d += 32) {
    const float hv = hrow[d];
#pragma unroll
    for (int e = 0; e < NEXP; ++e) acc[e] += hv * wr[d * NEXP + e];
  }
#pragma unroll
  for (int e = 0; e < NEXP; ++e)
    for (int off = 16; off > 0; off >>= 1)
      acc[e] += __shfl_xor(acc[e], off, 32);

  if (lane == 0) {
    const float* wrs = w_rskin + (size_t)l * 3 * NEXP;
    const float s0 = sprob[img * 3 + 0], s1 = sprob[img * 3 + 1], s2 = sprob[img * 3 + 2];
#pragma unroll
    for (int e = 0; e < NEXP; ++e)
      acc[e] += s0 * wrs[e] + s1 * wrs[NEXP + e] + s2 * wrs[2 * NEXP + e];
    float mx = acc[0];
#pragma unroll
    for (int e = 1; e < NEXP; ++e) mx = fmaxf(mx, acc[e]);
    float p[NEXP], s = 0.f;
#pragma unroll
    for (int e = 0; e < NEXP; ++e) { p[e] = __expf(acc[e] - mx); s += p[e]; }
    const float inv = 1.f / s;
#pragma unroll
    for (int e = 0; e < NEXP; ++e) p[e] *= inv;
    // top-2 (stable: first occurrence wins ties, like lax.top_k)
    int i0 = 0;
    for (int e = 1; e < NEXP; ++e) if (p[e] > p[i0]) i0 = e;
    int i1 = (i0 == 0) ? 1 : 0;
    for (int e = 0; e < NEXP; ++e) if (e != i0 && p[e] > p[i1]) i1 = e;
    const float norm = 1.f / (p[i0] + p[i1] + 1e-6f);
#pragma unroll
    for (int e = 0; e < NEXP; ++e) wfull[t * NEXP + e] = 0.f;
    wfull[t * NEXP + i0] = p[i0] * norm;
    wfull[t * NEXP + i1] = p[i1] * norm;
    float* ps = psum + (l * 8 + img) * NEXP;
    float* ms = msum + (l * 8 + img) * NEXP;
#pragma unroll
    for (int e = 0; e < NEXP; ++e) atomicAdd(&ps[e], p[e]);
    atomicAdd(&ms[i0], 1.f);
    atomicAdd(&ms[i1], 1.f);
  }
}

// ---------------------------------------------------------------------------
// Per-image column mean over 4096 tokens: out[img, d] = mean_t h[img, t, d]
// ---------------------------------------------------------------------------
__global__ void __launch_bounds__(256)
rowmean_kernel(const float* __restrict__ h, float* __restrict__ out)
{
  const int img = blockIdx.x >> 2;
  const int d = (blockIdx.x & 3) * 256 + threadIdx.x;
  const float* base = h + (size_t)img * TOKPI * DMODEL + d;
  float s = 0.f;
  for (int t = 0; t < TOKPI; ++t) s += base[(size_t)t * DMODEL];
  out[img * DMODEL + d] = s * (1.f / TOKPI);
}

// ---------------------------------------------------------------------------
// Skin classifier: relu(g@w_sc1+b1)@w_sc2+b2 -> logits out + softmax probs
// ---------------------------------------------------------------------------
__global__ void __launch_bounds__(64)
skin_kernel(const float* __restrict__ g, const float* __restrict__ w1,
            const float* __restrict__ b1, const float* __restrict__ w2,
            const float* __restrict__ b2, float* __restrict__ logits_out,
            float* __restrict__ sprob)
{
  const int img = blockIdx.x;
  const int j = threadIdx.x;
  __shared__ float hid[64];
  __shared__ float lg[3];
  const float* gi = g + img * DMODEL;
  float a = b1[j];
  for (int d = 0; d < DMODEL; ++d) a += gi[d] * w1[d * 64 + j];
  hid[j] = a > 0.f ? a : 0.f;
  __syncthreads();
  if (j < 3) {
    float s = b2[j];
    for (int k = 0; k < 64; ++k) s += hid[k] * w2[k * 3 + j];
    lg[j] = s;
    logits_out[img * 3 + j] = s;
  }
  __syncthreads();
  if (j == 0) {
    const float mx = fmaxf(lg[0], fmaxf(lg[1], lg[2]));
    const float e0 = __expf(lg[0] - mx), e1 = __expf(lg[1] - mx), e2 = __expf(lg[2] - mx);
    const float inv = 1.f / (e0 + e1 + e2);
    sprob[img * 3 + 0] = e0 * inv;
    sprob[img * 3 + 1] = e1 * inv;
    sprob[img * 3 + 2] = e2 * inv;
  }
}

// ---------------------------------------------------------------------------
// vision_proj = pooled @ w_out + b_out   ([8,1024]@[1024,4096])
// ---------------------------------------------------------------------------
__global__ void __launch_bounds__(256)
outproj_kernel(const float* __restrict__ pooled, const float* __restrict__ w_out,
               const float* __restrict__ b_out, float* __restrict__ vout)
{
  const int idx = blockIdx.x * 256 + threadIdx.x;   // 8*4096
  const int img = idx >> 12;
  const int o = idx & (OUTDIM - 1);
  const float* p = pooled + img * DMODEL;
  float s = b_out[o];
  for (int d = 0; d < DMODEL; ++d) s += p[d] * w_out[(size_t)d * OUTDIM + o];
  vout[idx] = s;
}

// ---------------------------------------------------------------------------
// aux = sum over L,img: E * sum_e (psum/T) * (msum/T)
// ---------------------------------------------------------------------------
__global__ void aux_kernel(const float* __restrict__ psum,
                           const float* __restrict__ msum,
                           float* __restrict__ out)
{
  if (threadIdx.x == 0 && blockIdx.x == 0) {
    const float invT = 1.f / TOKPI;
    float a = 0.f;
    for (int i = 0; i < NLAY * 8 * NEXP; ++i)
      a += (psum[i] * invT) * (msum[i] * invT);
    out[0] = a * (float)NEXP;
  }
}

__global__ void zero_kernel(float* __restrict__ p, int n)
{
  const int i = blockIdx.x * 256 + threadIdx.x;
  if (i < n) p[i] = 0.f;
}

// ---------------------------------------------------------------------------
extern "C" void kernel_launch(void* const* d_in, const int* in_sizes, int n_in,
                              void* d_out, int out_size, void* d_ws, size_t ws_size,
                              hipStream_t stream)
{
  const float* pixel   = (const float*)d_in[0];
  const float* w_in    = (const float*)d_in[2];
  const float* w_sc1   = (const float*)d_in[3];
  const float* b_sc1   = (const float*)d_in[4];
  const float* w_sc2   = (const float*)d_in[5];
  const float* b_sc2   = (const float*)d_in[6];
  const float* w_rimg  = (const float*)d_in[7];
  const float* w_rskin = (const float*)d_in[8];
  const float* ew1     = (const float*)d_in[9];
  const float* eb1     = (const float*)d_in[10];
  const float* ew2     = (const float*)d_in[11];
  const float* eb2     = (const float*)d_in[12];
  const float* w_out   = (const float*)d_in[13];
  const float* b_out   = (const float*)d_in[14];

  float* out = (float*)d_out;
  // output layout: pooled[8192] | vision_proj[32768] | aux[1] | skin_logits[24]
  float* out_pooled = out;
  float* out_vproj  = out + 8192;
  float* out_aux    = out + 40960;
  float* out_slog   = out + 40961;

  float* ws    = (float*)d_ws;
  float* h     = ws;                              // 33,554,432 f32
  float* h1w   = h + (size_t)NTOK * DMODEL;       // 16,777,216 f32
  float* wfull = h1w + (size_t)NTOK * NEXP * BN;  // 262,144 f32
  float* g     = wfull + (size_t)NTOK * NEXP;     // 8,192 f32
  float* sprob = g + 8 * DMODEL;                  // 32 f32 (24 used)
  float* psum  = sprob + 32;                      // 256 f32
  float* msum  = psum + NLAY * 8 * NEXP;          // 256 f32

  // zero aux accumulators (psum+msum contiguous)
  zero_kernel<<<2, 256, 0, stream>>>(psum, 2 * NLAY * 8 * NEXP);

  // h = X @ w_in
  gemm_in_kernel<<<dim3(DMODEL / 64, NTOK / 64), 128, 0, stream>>>(pixel, w_in, h);

  // skin head from pre-layer pooled features
  rowmean_kernel<<<32, 256, 0, stream>>>(h, g);
  skin_kernel<<<8, 64, 0, stream>>>(g, w_sc1, b_sc1, w_sc2, b_sc2, out_slog, sprob);

  for (int l = 0; l < NLAY; ++l) {
    router_kernel<<<NTOK / 8, 256, 0, stream>>>(h, w_rimg, w_rskin, sprob,
                                                wfull, psum, msum, l);
    gemm1_kernel<<<dim3(NEXP, NTOK / 64), 128, 0, stream>>>(h, ew1, eb1, wfull, h1w, l);
    gemm2_kernel<<<dim3(DMODEL / 64, NTOK / 64), 128, 0, stream>>>(h1w, ew2, eb2, wfull, h, l);
  }

  // pooled + out projection
  rowmean_kernel<<<32, 256, 0, stream>>>(h, out_pooled);
  outproj_kernel<<<(8 * OUTDIM) / 256, 256, 0, stream>>>(out_pooled, w_out, b_out, out_vproj);

  aux_kernel<<<1, 32, 0, stream>>>(psum, msum, out_aux);
}